// GCN3D_49898930045193
// MI455X (gfx1250) — compile-verified
//
#include <hip/hip_runtime.h>
#include <hip/hip_bf16.h>
#include <math.h>

// ---------------------------------------------------------------------------
// Types for WMMA f32 16x16x4 fragments (wave32):
//   A 16x4 f32  -> 64 vals / 32 lanes = 2 VGPRs  (v2f)
//   B 4x16 f32  -> 2 VGPRs                        (v2f)
//   C/D 16x16   -> 256 vals / 32 lanes = 8 VGPRs  (v8f)
// ---------------------------------------------------------------------------
typedef __attribute__((ext_vector_type(2))) float v2f;
typedef __attribute__((ext_vector_type(8))) float v8f;

// ---------------------------------------------------------------------------
// f32 GEMM via V_WMMA_F32_16X16X4_F32, 16x64 tile per wave (4 accumulators).
// C[M,N] = A[M,Kc] * B + bias
//   TRANSB==0 : B is row-major [Kc,N]
//   TRANSB==1 : B is W[N,Kc] row-major (computes A @ W^T)
// M multiple of 16, N multiple of 64, Kc multiple of 4.
// Fragment layouts per CDNA5 ISA 7.12.2:
//   A: row = lane&15; lanes<16 hold K=k+0,k+1; lanes>=16 hold K=k+2,k+3
//   B: col = lane&15; same K split across lane halves
//   D: VGPR r -> row = r + 8*(lane>>4), col = lane&15
// One A fragment feeds 4 WMMAs per K-step (operand reuse), and the v2f
// fragment loads are 8-byte aligned so they lower to global_load_b64.
// ---------------------------------------------------------------------------
template <int TRANSB>
__global__ void gemm_wmma_f32(const float* __restrict__ A,
                              const float* __restrict__ Bm,
                              const float* __restrict__ bias,
                              float* __restrict__ C,
                              int M, int N, int Kc) {
  const int wavesPerBlock = blockDim.x >> 5;
  const int wave = blockIdx.x * wavesPerBlock + (threadIdx.x >> 5);
  const int lane = threadIdx.x & 31;
  const int tilesN = N >> 6;                 // 64-wide N tiles
  const int tilesTotal = (M >> 4) * tilesN;
  if (wave >= tilesTotal) return;            // wave-uniform: EXEC stays all-ones

  const int tm = (wave / tilesN) << 4;
  const int tn = (wave % tilesN) << 6;
  const int half = lane >> 4;                // 0 or 1
  const int l15  = lane & 15;

  v8f acc0 = {0.f,0.f,0.f,0.f,0.f,0.f,0.f,0.f};
  v8f acc1 = acc0, acc2 = acc0, acc3 = acc0;

  const float* arow = A + (size_t)(tm + l15) * Kc;
  const float* wrow0 = nullptr; const float* wrow1 = nullptr;
  const float* wrow2 = nullptr; const float* wrow3 = nullptr;
  if (TRANSB) {
    wrow0 = Bm + (size_t)(tn +  0 + l15) * Kc;
    wrow1 = Bm + (size_t)(tn + 16 + l15) * Kc;
    wrow2 = Bm + (size_t)(tn + 32 + l15) * Kc;
    wrow3 = Bm + (size_t)(tn + 48 + l15) * Kc;
  }

  for (int k = 0; k < Kc; k += 4) {
    const int ka = k + 2 * half;             // even -> 8B-aligned v2f loads
    const v2f a = *(const v2f*)(arow + ka);
    v2f b0, b1, b2, b3;
    if (TRANSB) {
      b0 = *(const v2f*)(wrow0 + ka);
      b1 = *(const v2f*)(wrow1 + ka);
      b2 = *(const v2f*)(wrow2 + ka);
      b3 = *(const v2f*)(wrow3 + ka);
    } else {
      const float* br0 = Bm + (size_t)ka * N + tn + l15;
      const float* br1 = Bm + (size_t)(ka + 1) * N + tn + l15;
      b0.x = br0[0];  b0.y = br1[0];
      b1.x = br0[16]; b1.y = br1[16];
      b2.x = br0[32]; b2.y = br1[32];
      b3.x = br0[48]; b3.y = br1[48];
    }
    // speculative prefetch ~8 K-steps ahead (lowers to global_prefetch_b8)
    if (k + 32 < Kc) {
      __builtin_prefetch(arow + ka + 32, 0, 0);
      if (TRANSB) __builtin_prefetch(wrow0 + ka + 32, 0, 0);
      else        __builtin_prefetch(Bm + (size_t)(ka + 32) * N + tn + l15, 0, 0);
    }
    acc0 = __builtin_amdgcn_wmma_f32_16x16x4_f32(false, a, false, b0, (short)0, acc0, false, false);
    acc1 = __builtin_amdgcn_wmma_f32_16x16x4_f32(false, a, false, b1, (short)0, acc1, false, false);
    acc2 = __builtin_amdgcn_wmma_f32_16x16x4_f32(false, a, false, b2, (short)0, acc2, false, false);
    acc3 = __builtin_amdgcn_wmma_f32_16x16x4_f32(false, a, false, b3, (short)0, acc3, false, false);
  }

#pragma unroll
  for (int j = 0; j < 4; ++j) {
    const v8f* accs[4] = {&acc0, &acc1, &acc2, &acc3};
    const v8f av = *accs[j];
    const int col = tn + 16 * j + l15;
    const float bv = bias ? bias[col] : 0.f;
#pragma unroll
    for (int r = 0; r < 8; ++r) {
      const int row = tm + r + 8 * half;
      C[(size_t)row * N + col] = av[r] + bv;
    }
  }
}

// ---------------------------------------------------------------------------
// KNN: per query point, K smallest squared distances (self excluded),
// insertion sort of length Kn (<=20). Optional deterministic subset select
// (stand-in for jax.random.permutation, unverifiable compile-only).
// ---------------------------------------------------------------------------
__global__ void knn_kernel(const float* __restrict__ verts, int* __restrict__ idx,
                           int Vn, int Kn, int nQuery, int useSel, unsigned mult) {
  const int b = blockIdx.y;
  const int i = blockIdx.x * blockDim.x + threadIdx.x;
  if (i >= nQuery) return;
  const int qi = useSel ? (int)(((unsigned)i * mult) & (unsigned)(Vn - 1)) : i;
  const float* vb = verts + (size_t)b * Vn * 3;
  const float qx = vb[qi * 3 + 0], qy = vb[qi * 3 + 1], qz = vb[qi * 3 + 2];

  float bd[20];
  int   bi[20];
  for (int t = 0; t < Kn; ++t) { bd[t] = 3.4e38f; bi[t] = 0; }

  for (int j = 0; j < Vn; ++j) {
    if (j == qi) continue;
    const float dx = vb[j * 3 + 0] - qx;
    const float dy = vb[j * 3 + 1] - qy;
    const float dz = vb[j * 3 + 2] - qz;
    const float d = dx * dx + dy * dy + dz * dz;
    if (d < bd[Kn - 1]) {
      int p = Kn - 1;
      while (p > 0 && bd[p - 1] > d) {
        bd[p] = bd[p - 1]; bi[p] = bi[p - 1]; --p;
      }
      bd[p] = d; bi[p] = j;
    }
  }
  int* op = idx + ((size_t)b * nQuery + i) * Kn;
  for (int t = 0; t < Kn; ++t) op[t] = bi[t];
}

// Normalized neighbor directions nd[b,i,k,3].
__global__ void dirs_kernel(const float* __restrict__ verts,
                            const int* __restrict__ idx,
                            float* __restrict__ nd,
                            int Vn, int Kn, int total) {
  const int t = blockIdx.x * blockDim.x + threadIdx.x;
  if (t >= total) return;  // t enumerates (b,i,k)
  const int r = t / Kn;            // b*Vn + i
  const int i = r % Vn;
  const int b = r / Vn;
  const float* vb = verts + (size_t)b * Vn * 3;
  const int j = idx[t];
  float dx = vb[j * 3 + 0] - vb[i * 3 + 0];
  float dy = vb[j * 3 + 1] - vb[i * 3 + 1];
  float dz = vb[j * 3 + 2] - vb[i * 3 + 2];
  float n = fmaxf(sqrtf(dx * dx + dy * dy + dz * dz), 1e-12f);
  nd[(size_t)t * 3 + 0] = dx / n;
  nd[(size_t)t * 3 + 1] = dy / n;
  nd[(size_t)t * 3 + 2] = dz / n;
}

// conv_surface fused with LayerNorm + ReLU. 32 output channels.
__global__ void conv_surface_ln_kernel(const float* __restrict__ nd,
                                       const float* __restrict__ d0,
                                       const float* __restrict__ g,
                                       const float* __restrict__ be,
                                       float* __restrict__ out,
                                       int Kn, int total) {
  const int t = blockIdx.x * blockDim.x + threadIdx.x;
  if (t >= total) return;  // t = b*Vn + i
  const float* ndp = nd + (size_t)t * Kn * 3;
  float vals[32];
  for (int m = 0; m < 32; ++m) {
    float dx = d0[m], dy = d0[32 + m], dz = d0[64 + m];
    float n = fmaxf(sqrtf(dx * dx + dy * dy + dz * dz), 1e-12f);
    dx /= n; dy /= n; dz /= n;
    float best = -3.4e38f;
    for (int k = 0; k < Kn; ++k) {
      float th = fmaxf(dx * ndp[3 * k] + dy * ndp[3 * k + 1] + dz * ndp[3 * k + 2], 0.f);
      best = fmaxf(best, th);
    }
    vals[m] = best;
  }
  float mean = 0.f;
  for (int m = 0; m < 32; ++m) mean += vals[m];
  mean *= (1.f / 32.f);
  float var = 0.f;
  for (int m = 0; m < 32; ++m) { float d = vals[m] - mean; var += d * d; }
  var *= (1.f / 32.f);
  const float inv = rsqrtf(var + 1e-6f);
  for (int m = 0; m < 32; ++m)
    out[(size_t)t * 32 + m] = fmaxf((vals[m] - mean) * inv * g[m] + be[m], 0.f);
}

// conv_layer combine: out = center + max_k( relu(nd . dir_c) * support )
// fo has 2*OC columns: [center | support].
__global__ void conv_combine_kernel(const float* __restrict__ fo,
                                    const int* __restrict__ idx,
                                    const float* __restrict__ nd,
                                    const float* __restrict__ dir,
                                    float* __restrict__ out,
                                    int Vn, int Kn, int OC, int total) {
  const int t = blockIdx.x * blockDim.x + threadIdx.x;
  if (t >= total) return;  // t = (b*Vn + i)*OC + c
  const int c = t % OC;
  const int r = t / OC;    // b*Vn + i
  const int b = r / Vn;
  float dx = dir[c], dy = dir[OC + c], dz = dir[2 * OC + c];
  float n = fmaxf(sqrtf(dx * dx + dy * dy + dz * dz), 1e-12f);
  dx /= n; dy /= n; dz /= n;
  const int* ip = idx + (size_t)r * Kn;
  const float* ndp = nd + (size_t)r * Kn * 3;
  const size_t ld = (size_t)2 * OC;
  float best = -3.4e38f;
  for (int k = 0; k < Kn; ++k) {
    const float th = fmaxf(dx * ndp[3 * k] + dy * ndp[3 * k + 1] + dz * ndp[3 * k + 2], 0.f);
    const float sup = fo[((size_t)b * Vn + ip[k]) * ld + OC + c];
    best = fmaxf(best, th * sup);
  }
  out[t] = fo[(size_t)r * ld + c] + best;
}

// LayerNorm over last dim + ReLU.
__global__ void ln_relu_kernel(const float* __restrict__ x,
                               const float* __restrict__ g,
                               const float* __restrict__ be,
                               float* __restrict__ y, int rows, int C) {
  const int r = blockIdx.x * blockDim.x + threadIdx.x;
  if (r >= rows) return;
  const float* xr = x + (size_t)r * C;
  float mean = 0.f;
  for (int c = 0; c < C; ++c) mean += xr[c];
  mean /= (float)C;
  float var = 0.f;
  for (int c = 0; c < C; ++c) { float d = xr[c] - mean; var += d * d; }
  var /= (float)C;
  const float inv = rsqrtf(var + 1e-6f);
  float* yr = y + (size_t)r * C;
  for (int c = 0; c < C; ++c)
    yr[c] = fmaxf((xr[c] - mean) * inv * g[c] + be[c], 0.f);
}

// Select subset of vertex rows (deterministic odd-multiplier bijection).
__global__ void select_verts_kernel(const float* __restrict__ src,
                                    float* __restrict__ dst,
                                    int Vn, int nsel, unsigned mult, int total) {
  const int t = blockIdx.x * blockDim.x + threadIdx.x;
  if (t >= total) return;  // t = b*nsel + i
  const int i = t % nsel;
  const int b = t / nsel;
  const int s = (int)(((unsigned)i * mult) & (unsigned)(Vn - 1));
  dst[(size_t)t * 3 + 0] = src[((size_t)b * Vn + s) * 3 + 0];
  dst[(size_t)t * 3 + 1] = src[((size_t)b * Vn + s) * 3 + 1];
  dst[(size_t)t * 3 + 2] = src[((size_t)b * Vn + s) * 3 + 2];
}

// Max-pool of features over Kn neighbor indices.
__global__ void pool_max_kernel(const float* __restrict__ fm,
                                const int* __restrict__ idxk,
                                float* __restrict__ out,
                                int Vn, int C, int nsel, int Kn, int total) {
  const int t = blockIdx.x * blockDim.x + threadIdx.x;
  if (t >= total) return;  // t = (b*nsel + i)*C + c
  const int c = t % C;
  const int r = t / C;
  const int b = r / nsel;
  const int* ip = idxk + (size_t)r * Kn;
  float best = -3.4e38f;
  for (int k = 0; k < Kn; ++k)
    best = fmaxf(best, fm[((size_t)b * Vn + ip[k]) * C + c]);
  out[t] = best;
}

// token (1,64,1024,1) -> tok[b,n,e] (B,1024,64)
__global__ void tok_kernel(const float* __restrict__ token,
                           float* __restrict__ tok, int total) {
  const int t = blockIdx.x * blockDim.x + threadIdx.x;
  if (t >= total) return;  // t = (b*1024 + n)*64 + e
  const int e = t % 64;
  const int n = (t / 64) % 1024;
  tok[t] = token[e * 1024 + n];
}

// Flash-style single-pass attention: one thread per (b,h,q).
template <int HD>
__global__ void attn_kernel(const float* __restrict__ Q,
                            const float* __restrict__ Kx,
                            const float* __restrict__ Vx,
                            float* __restrict__ O,
                            int NQ, int NK, int H, int total) {
  const int t = blockIdx.x * blockDim.x + threadIdx.x;
  if (t >= total) return;  // t = (b*H + h)*NQ + q
  const int q = t % NQ;
  const int h = (t / NQ) % H;
  const int b = t / (NQ * H);
  const int E = H * HD;
  const float* qp = Q + ((size_t)b * NQ + q) * E + h * HD;
  float qv[HD];
#pragma unroll
  for (int d = 0; d < HD; ++d) qv[d] = qp[d];
  float m = -3.4e38f, s = 0.f;
  float acc[HD];
#pragma unroll
  for (int d = 0; d < HD; ++d) acc[d] = 0.f;
  const float scale = rsqrtf((float)HD);
  for (int j = 0; j < NK; ++j) {
    const float* kp = Kx + ((size_t)b * NK + j) * E + h * HD;
    float dot = 0.f;
#pragma unroll
    for (int d = 0; d < HD; ++d) dot += qv[d] * kp[d];
    dot *= scale;
    const float mn = fmaxf(m, dot);
    const float corr = expf(m - mn);
    const float w = expf(dot - mn);
    s = s * corr + w;
    const float* vp = Vx + ((size_t)b * NK + j) * E + h * HD;
#pragma unroll
    for (int d = 0; d < HD; ++d) acc[d] = acc[d] * corr + w * vp[d];
    m = mn;
  }
  float* op = O + ((size_t)b * NQ + q) * E + h * HD;
  const float invs = 1.f / s;
#pragma unroll
  for (int d = 0; d < HD; ++d) op[d] = acc[d] * invs;
}

// Elementwise residual add.
__global__ void add_kernel(const float* __restrict__ a,
                           const float* __restrict__ b,
                           float* __restrict__ o, int n) {
  const int t = blockIdx.x * blockDim.x + threadIdx.x;
  if (t < n) o[t] = a[t] + b[t];
}

// BatchNorm(train) per channel over rows, then ReLU, in place. One block / ch.
__global__ void bn_relu_kernel(float* __restrict__ x,
                               const float* __restrict__ g,
                               const float* __restrict__ be,
                               int rows, int C) {
  const int c = blockIdx.x;
  __shared__ float ssum[256];
  __shared__ float ssq[256];
  float s = 0.f, sq = 0.f;
  for (int r = threadIdx.x; r < rows; r += blockDim.x) {
    const float v = x[(size_t)r * C + c];
    s += v; sq += v * v;
  }
  ssum[threadIdx.x] = s; ssq[threadIdx.x] = sq;
  __syncthreads();
  for (int off = blockDim.x >> 1; off > 0; off >>= 1) {
    if ((int)threadIdx.x < off) {
      ssum[threadIdx.x] += ssum[threadIdx.x + off];
      ssq[threadIdx.x]  += ssq[threadIdx.x + off];
    }
    __syncthreads();
  }
  const float mean = ssum[0] / (float)rows;
  const float var  = ssq[0] / (float)rows - mean * mean;
  const float inv  = rsqrtf(var + 1e-5f);
  const float gg = g[c], bb = be[c];
  for (int r = threadIdx.x; r < rows; r += blockDim.x) {
    const float v = (x[(size_t)r * C + c] - mean) * inv * gg + bb;
    x[(size_t)r * C + c] = fmaxf(v, 0.f);
  }
}

// (b*N+n, c) row-major -> (b, c, n) output layout.
__global__ void transpose_out_kernel(const float* __restrict__ src,
                                     float* __restrict__ dst,
                                     int N, int C, int total) {
  const int t = blockIdx.x * blockDim.x + threadIdx.x;
  if (t >= total) return;  // t = b*C*N + c*N + n
  const int n = t % N;
  const int r = t / N;
  const int c = r % C;
  const int b = r / C;
  dst[t] = src[((size_t)b * N + n) * C + c];
}

// ---------------------------------------------------------------------------
extern "C" void kernel_launch(void* const* d_in, const int* in_sizes, int n_in,
                              void* d_out, int out_size, void* d_ws, size_t ws_size,
                              hipStream_t stream) {
  (void)in_sizes; (void)n_in; (void)out_size; (void)ws_size;
  const int Bn = 4, V0 = 4096, V1 = 1024, V2 = 256, KN = 20;
  const int rows0 = Bn * V0;   // 16384
  const int rows1 = Bn * V1;   // 4096
  const unsigned MULT = 2654435761u;  // odd -> bijection mod 2^n

  // -------- inputs (setup_inputs dict order) --------
  const float* vertices = (const float*)d_in[0];   // (B,1,V,3)
  const float* d0       = (const float*)d_in[1];
  const float* g0       = (const float*)d_in[2];
  const float* be0      = (const float*)d_in[3];
  const float* w1       = (const float*)d_in[4];
  const float* b1       = (const float*)d_in[5];
  const float* dir1     = (const float*)d_in[6];
  const float* g1       = (const float*)d_in[7];
  const float* be1      = (const float*)d_in[8];
  const float* w2       = (const float*)d_in[9];
  const float* b2       = (const float*)d_in[10];
  const float* dir2     = (const float*)d_in[11];
  const float* g2       = (const float*)d_in[12];
  const float* be2      = (const float*)d_in[13];
  const float* w3       = (const float*)d_in[14];
  const float* b3       = (const float*)d_in[15];
  const float* dir3     = (const float*)d_in[16];
  const float* g3       = (const float*)d_in[17];
  const float* be3      = (const float*)d_in[18];
  const float* token    = (const float*)d_in[19];
  const float* a1_in_w  = (const float*)d_in[20];
  const float* a1_in_b  = (const float*)d_in[21];
  const float* a1_out_w = (const float*)d_in[22];
  const float* a1_out_b = (const float*)d_in[23];
  const float* ffn1_w   = (const float*)d_in[24];
  const float* ffn1_b   = (const float*)d_in[25];
  const float* bg       = (const float*)d_in[26];
  const float* bb       = (const float*)d_in[27];
  const float* ffn2_w   = (const float*)d_in[28];
  const float* ffn2_b   = (const float*)d_in[29];
  const float* a2_in_w  = (const float*)d_in[30];
  const float* a2_in_b  = (const float*)d_in[31];
  const float* a2_out_w = (const float*)d_in[32];
  const float* a2_out_b = (const float*)d_in[33];
  const float* w7       = (const float*)d_in[34];
  const float* b7       = (const float*)d_in[35];
  const float* dir7     = (const float*)d_in[36];
  const float* g4       = (const float*)d_in[37];
  const float* be4      = (const float*)d_in[38];
  const float* c8_w     = (const float*)d_in[39];
  const float* c8_b     = (const float*)d_in[40];

  // -------- workspace bump allocator --------
  char* wsp = (char*)d_ws;
  size_t off = 0;
  auto alloc = [&](size_t nbytes) -> void* {
    void* p = wsp + off;
    off += (nbytes + 255) & ~(size_t)255;
    return p;
  };
  int*   idx0    = (int*)  alloc((size_t)Bn * V0 * KN * 4);
  float* nd0     = (float*)alloc((size_t)Bn * V0 * KN * 3 * 4);
  float* fm0     = (float*)alloc((size_t)rows0 * 32 * 4);
  float* fo1     = (float*)alloc((size_t)rows0 * 128 * 4);
  float* fm1raw  = (float*)alloc((size_t)rows0 * 64 * 4);
  float* fm1     = (float*)alloc((size_t)rows0 * 64 * 4);
  float* v1      = (float*)alloc((size_t)Bn * V1 * 3 * 4);
  int*   idx4a   = (int*)  alloc((size_t)Bn * V1 * 4 * 4);
  float* fm1p    = (float*)alloc((size_t)rows1 * 64 * 4);
  int*   idx1    = (int*)  alloc((size_t)Bn * V1 * KN * 4);
  float* nd1     = (float*)alloc((size_t)Bn * V1 * KN * 3 * 4);
  float* fo2     = (float*)alloc((size_t)rows1 * 256 * 4);
  float* fm2raw  = (float*)alloc((size_t)rows1 * 128 * 4);
  float* fm2     = (float*)alloc((size_t)rows1 * 128 * 4);
  float* fo3     = (float*)alloc((size_t)rows1 * 512 * 4);
  float* fm3raw  = (float*)alloc((size_t)rows1 * 256 * 4);
  float* fm3     = (float*)alloc((size_t)rows1 * 256 * 4);
  int*   idx4b   = (int*)  alloc((size_t)Bn * V2 * 4 * 4);
  float* fm3p    = (float*)alloc((size_t)Bn * V2 * 256 * 4);
  float* tok     = (float*)alloc((size_t)rows1 * 64 * 4);
  float* qh      = (float*)alloc((size_t)rows1 * 64 * 4);
  float* kh      = (float*)alloc((size_t)rows1 * 64 * 4);
  float* vh      = (float*)alloc((size_t)rows1 * 64 * 4);
  float* attno   = (float*)alloc((size_t)rows1 * 64 * 4);
  float* mha1o   = (float*)alloc((size_t)rows1 * 64 * 4);
  float* query   = (float*)alloc((size_t)rows1 * 64 * 4);
  float* y1      = (float*)alloc((size_t)rows1 * 256 * 4);
  float* query2  = (float*)alloc((size_t)rows1 * 256 * 4);
  float* q2      = (float*)alloc((size_t)rows1 * 256 * 4);
  float* k2      = (float*)alloc((size_t)Bn * V2 * 256 * 4);
  float* v2b     = (float*)alloc((size_t)Bn * V2 * 256 * 4);
  float* attno2  = (float*)alloc((size_t)rows1 * 256 * 4);
  float* mha2o   = (float*)alloc((size_t)rows1 * 256 * 4);
  float* cross   = (float*)alloc((size_t)rows1 * 256 * 4);
  float* fo7     = (float*)alloc((size_t)rows1 * 2048 * 4);
  float* fmfraw  = (float*)alloc((size_t)rows1 * 1024 * 4);
  float* fmf     = (float*)alloc((size_t)rows1 * 1024 * 4);
  float* outtmp  = (float*)alloc((size_t)rows1 * 256 * 4);

  auto blocks1d = [](int n) { return dim3((unsigned)((n + 255) / 256)); };
  auto gemm = [&](const float* A, const float* Bm, const float* bias, float* C,
                  int M, int N, int Kc) {
    int tiles = (M / 16) * (N / 64);
    gemm_wmma_f32<0><<<dim3((tiles + 3) / 4), dim3(128), 0, stream>>>(A, Bm, bias, C, M, N, Kc);
  };
  auto gemmT = [&](const float* A, const float* W, const float* bias, float* C,
                   int M, int N, int Kc) {
    int tiles = (M / 16) * (N / 64);
    gemm_wmma_f32<1><<<dim3((tiles + 3) / 4), dim3(128), 0, stream>>>(A, W, bias, C, M, N, Kc);
  };

  // ---- stage 1: knn graph on v0, surface conv, conv_layer 1 ----
  knn_kernel<<<dim3((V0 + 127) / 128, Bn), dim3(128), 0, stream>>>(
      vertices, idx0, V0, KN, V0, 0, MULT);
  {
    int total = Bn * V0 * KN;
    dirs_kernel<<<blocks1d(total), 256, 0, stream>>>(vertices, idx0, nd0, V0, KN, total);
  }
  conv_surface_ln_kernel<<<blocks1d(rows0), 256, 0, stream>>>(nd0, d0, g0, be0, fm0, KN, rows0);
  gemm(fm0, w1, b1, fo1, rows0, 128, 32);
  {
    int total = rows0 * 64;
    conv_combine_kernel<<<blocks1d(total), 256, 0, stream>>>(fo1, idx0, nd0, dir1, fm1raw, V0, KN, 64, total);
  }
  ln_relu_kernel<<<blocks1d(rows0), 256, 0, stream>>>(fm1raw, g1, be1, fm1, rows0, 64);

  // ---- pool 4096 -> 1024 ----
  select_verts_kernel<<<blocks1d(Bn * V1), 256, 0, stream>>>(vertices, v1, V0, V1, MULT, Bn * V1);
  knn_kernel<<<dim3((V1 + 127) / 128, Bn), dim3(128), 0, stream>>>(
      vertices, idx4a, V0, 4, V1, 1, MULT);
  {
    int total = rows1 * 64;
    pool_max_kernel<<<blocks1d(total), 256, 0, stream>>>(fm1, idx4a, fm1p, V0, 64, V1, 4, total);
  }

  // ---- stage 2: knn on v1, conv_layer 2 and 3 ----
  knn_kernel<<<dim3((V1 + 127) / 128, Bn), dim3(128), 0, stream>>>(v1, idx1, V1, KN, V1, 0, MULT);
  {
    int total = Bn * V1 * KN;
    dirs_kernel<<<blocks1d(total), 256, 0, stream>>>(v1, idx1, nd1, V1, KN, total);
  }
  gemm(fm1p, w2, b2, fo2, rows1, 256, 64);
  {
    int total = rows1 * 128;
    conv_combine_kernel<<<blocks1d(total), 256, 0, stream>>>(fo2, idx1, nd1, dir2, fm2raw, V1, KN, 128, total);
  }
  ln_relu_kernel<<<blocks1d(rows1), 256, 0, stream>>>(fm2raw, g2, be2, fm2, rows1, 128);
  gemm(fm2, w3, b3, fo3, rows1, 512, 128);
  {
    int total = rows1 * 256;
    conv_combine_kernel<<<blocks1d(total), 256, 0, stream>>>(fo3, idx1, nd1, dir3, fm3raw, V1, KN, 256, total);
  }
  ln_relu_kernel<<<blocks1d(rows1), 256, 0, stream>>>(fm3raw, g3, be3, fm3, rows1, 256);

  // ---- pool 1024 -> 256 (features only) ----
  knn_kernel<<<dim3((V2 + 127) / 128, Bn), dim3(128), 0, stream>>>(v1, idx4b, V1, 4, V2, 1, MULT);
  {
    int total = Bn * V2 * 256;
    pool_max_kernel<<<blocks1d(total), 256, 0, stream>>>(fm3, idx4b, fm3p, V1, 256, V2, 4, total);
  }

  // ---- MHA 1: tok queries, fm1p keys/values (E=64, h=4, hd=16) ----
  tok_kernel<<<blocks1d(rows1 * 64), 256, 0, stream>>>(token, tok, rows1 * 64);
  gemmT(tok,  a1_in_w,            a1_in_b,       qh, rows1, 64, 64);
  gemmT(fm1p, a1_in_w + 64 * 64,  a1_in_b + 64,  kh, rows1, 64, 64);
  gemmT(fm1p, a1_in_w + 128 * 64, a1_in_b + 128, vh, rows1, 64, 64);
  attn_kernel<16><<<blocks1d(Bn * 4 * V1), 256, 0, stream>>>(qh, kh, vh, attno, V1, V1, 4, Bn * 4 * V1);
  gemmT(attno, a1_out_w, a1_out_b, mha1o, rows1, 64, 64);
  add_kernel<<<blocks1d(rows1 * 64), 256, 0, stream>>>(mha1o, fm1p, query, rows1 * 64);

  // ---- FFN with BN ----
  gemmT(query, ffn1_w, ffn1_b, y1, rows1, 256, 64);
  bn_relu_kernel<<<dim3(256), dim3(256), 0, stream>>>(y1, bg, bb, rows1, 256);
  gemmT(y1, ffn2_w, ffn2_b, query2, rows1, 256, 256);

  // ---- MHA 2: query2 queries, fm3p keys/values (E=256, h=4, hd=64) ----
  gemmT(query2, a2_in_w,             a2_in_b,       q2,  rows1,    256, 256);
  gemmT(fm3p,   a2_in_w + 256 * 256, a2_in_b + 256, k2,  Bn * V2,  256, 256);
  gemmT(fm3p,   a2_in_w + 512 * 256, a2_in_b + 512, v2b, Bn * V2,  256, 256);
  attn_kernel<64><<<blocks1d(Bn * 4 * V1), 256, 0, stream>>>(q2, k2, v2b, attno2, V1, V2, 4, Bn * 4 * V1);
  gemmT(attno2, a2_out_w, a2_out_b, mha2o, rows1, 256, 256);
  add_kernel<<<blocks1d(rows1 * 256), 256, 0, stream>>>(mha2o, query2, cross, rows1 * 256);

  // ---- final conv_layer (out_c=1024) + 1x1 conv ----
  gemm(cross, w7, b7, fo7, rows1, 2048, 256);
  {
    int total = rows1 * 1024;
    conv_combine_kernel<<<blocks1d(total), 256, 0, stream>>>(fo7, idx1, nd1, dir7, fmfraw, V1, KN, 1024, total);
  }
  ln_relu_kernel<<<blocks1d(rows1), 256, 0, stream>>>(fmfraw, g4, be4, fmf, rows1, 1024);
  gemmT(fmf, c8_w, c8_b, outtmp, rows1, 256, 1024);
  {
    int total = Bn * 256 * V1;
    transpose_out_kernel<<<blocks1d(total), 256, 0, stream>>>(outtmp, (float*)d_out, V1, 256, total);
  }
}